// QuantumAttention_81003083202991
// MI455X (gfx1250) — compile-verified
//
#include <hip/hip_runtime.h>

// QuantumAttention collapses algebraically:
//   attn  == 1/S everywhere (constant-score softmax, exactly 2^-11 in fp32)
//   out[b,i,:] == (mean_s value[b,s,:]) @ Wv + bv, then @ Wo + bo  (same for all i)
// Cost is dominated by the 553.6 MB output store -> ~24 us HBM floor on MI455X.
// WMMA f32 16x16x4 used for the two small GEMMs; M=2 padded to 16 via a padded
// workspace buffer so the inner loop has zero control flow (no exec masking).

typedef __attribute__((ext_vector_type(2))) float v2f;
typedef __attribute__((ext_vector_type(4))) float f4;
typedef __attribute__((ext_vector_type(8))) float v8f;

#define B_ 2
#define S_ 2048
#define E_ 1024

// ---------- Stage 1a: partial column sums of `value` over S chunks ----------
// grid (8, 16), block 256. col = b*E + e  (0..2047). Deterministic (no atomics).
__global__ void qa_colsum_part(const float* __restrict__ value,
                               float* __restrict__ partial) {
    int col   = blockIdx.x * 256 + threadIdx.x;   // 0..2047
    int chunk = blockIdx.y;                        // 0..15, each covers 128 rows of S
    int b = col >> 10, e = col & 1023;
    const float* p = value + (size_t)b * S_ * E_ + (size_t)chunk * 128 * E_ + e;
    float s = 0.f;
#pragma unroll 4
    for (int i = 0; i < 128; ++i) s += p[(size_t)i * E_];
    partial[chunk * 2048 + col] = s;
}

// ---------- Stage 1b: fold partials -> padded [16 x 1024] column mean ----------
// Rows 0..1 = per-batch mean; rows 2..15 = 0 (M-padding for the 16x16 WMMA tile).
__global__ void qa_colsum_final(const float* __restrict__ partial,
                                float* __restrict__ cmean_pad) {
    int idx = blockIdx.x * 256 + threadIdx.x;     // 64 blocks x 256 = 16384
    float v = 0.f;
    if (idx < 2048) {
        float s = 0.f;
#pragma unroll
        for (int c = 0; c < 16; ++c) s += partial[c * 2048 + idx];
        v = s * (1.0f / (float)S_);
    }
    cmean_pad[idx] = v;
}

// ---------- Stage 2: [16 x 1024] @ [1024 x 1024] + bias via V_WMMA_F32_16X16X4_F32 ----
// One wave per 16-wide N tile (64 waves). Branch-free inner loop:
// ISA 32-bit A 16x4 layout: lanes 0-15 -> VGPR0=K0,VGPR1=K1 ; lanes 16-31 -> K2,K3.
// B 4x16 mirrored row-striping. C/D: VGPR r = rows {r, r+8*khalf}.
// All 16 output rows are stored (rows >= 2 are dead padding in the workspace);
// each output row depends only on its own A row, so padding rows never taint rows 0-1.
__global__ __launch_bounds__(32) void qa_gemm16_wmma(
        const float* __restrict__ A,    // [16,1024] padded
        const float* __restrict__ W,    // [1024,1024] row-major
        const float* __restrict__ bias, // [1024]
        float* __restrict__ out) {      // [16,1024] padded (rows 0..1 meaningful)
    const int lane  = threadIdx.x;
    const int n0    = blockIdx.x * 16;
    const int m     = lane & 15;       // A row (lanes 0-15) / also N column index
    const int khalf = lane >> 4;       // 0 -> K{0,1}, 1 -> K{2,3}
    const float* arow = A + (size_t)m * E_;
    const float* wcol = W + n0 + m;    // column n = m for this lane

    v8f acc = {};
#pragma unroll 4
    for (int k0 = 0; k0 < E_; k0 += 4) {
        const int kb = k0 + 2 * khalf;
        v2f a = *(const v2f*)(arow + kb);            // 8B-aligned b64 load
        v2f b;
        b.x = wcol[(size_t)kb * E_];
        b.y = wcol[(size_t)(kb + 1) * E_];
        acc = __builtin_amdgcn_wmma_f32_16x16x4_f32(
                /*neg_a=*/false, a, /*neg_b=*/false, b,
                /*c_mod=*/(short)0, acc, /*reuse_a=*/false, /*reuse_b=*/false);
    }
    const float bb = bias[n0 + m];
#pragma unroll
    for (int r = 0; r < 8; ++r)
        out[(size_t)(r + 8 * khalf) * E_ + n0 + m] = acc[r] + bb;
}

// ---------- Stage 3a: broadcast f[b,:] across all S rows of `out` ----------
// 1,048,576 float4 NT stores (16.8 MB).
__global__ void qa_bcast_out(const float* __restrict__ f, float* __restrict__ out) {
    int t  = blockIdx.x * 256 + threadIdx.x;  // 0..1048575
    int e4 = t & 255;                          // E/4 = 256 float4 per row
    int bs = t >> 8;                           // b*S + s
    int b  = bs >> 11;                         // S = 2048
    f4 v = ((const f4*)(f + (size_t)b * E_))[e4];   // hot in L2, heavy reuse
    __builtin_nontemporal_store(v, ((f4*)out) + t);
}

// ---------- Stage 3b: fill attn with the exact constant 1/2048 ----------
// 33,554,432 float4 NT stores (537 MB) — this kernel sets the runtime.
__global__ void qa_attn_fill(float* __restrict__ attn) {
    const float c = 0.00048828125f;            // exactly 1/2048
    const f4 v = {c, c, c, c};
    f4* p = (f4*)attn;
    const size_t base   = (size_t)blockIdx.x * 256 + threadIdx.x;
    const size_t stride = (size_t)32768 * 256; // gridDim.x * blockDim.x
#pragma unroll
    for (int i = 0; i < 4; ++i)
        __builtin_nontemporal_store(v, p + base + (size_t)i * stride);
}

extern "C" void kernel_launch(void* const* d_in, const int* in_sizes, int n_in,
                              void* d_out, int out_size, void* d_ws, size_t ws_size,
                              hipStream_t stream) {
    (void)in_sizes; (void)n_in; (void)out_size; (void)ws_size;
    // setup_inputs order: query(0) key(1) value(2) Wq(3) bq(4) Wk(5) bk(6)
    //                     Wv(7) bv(8) Wo(9) bo(10) — all float32.
    const float* value = (const float*)d_in[2];
    const float* Wv    = (const float*)d_in[7];
    const float* bv    = (const float*)d_in[8];
    const float* Wo    = (const float*)d_in[9];
    const float* bo    = (const float*)d_in[10];
    float* out = (float*)d_out;

    float* ws        = (float*)d_ws;
    float* partial   = ws;             // [16][2048]   128 KB
    float* cmean_pad = ws + 32768;     // [16][1024]    64 KB (rows 2..15 zero)
    float* mbuf_pad  = ws + 49152;     // [16][1024]    64 KB (rows 2..15 dead)
    float* fbuf_pad  = ws + 65536;     // [16][1024]    64 KB (rows 2..15 dead)

    qa_colsum_part <<<dim3(8, 16), 256, 0, stream>>>(value, partial);
    qa_colsum_final<<<64,         256, 0, stream>>>(partial, cmean_pad);
    qa_gemm16_wmma <<<64,          32, 0, stream>>>(cmean_pad, Wv, bv, mbuf_pad);
    qa_gemm16_wmma <<<64,          32, 0, stream>>>(mbuf_pad,  Wo, bo, fbuf_pad);
    qa_bcast_out   <<<4096,       256, 0, stream>>>(fbuf_pad, out);
    qa_attn_fill   <<<32768,      256, 0, stream>>>(out + (size_t)B_ * S_ * E_);
}